// GATv2_1838246002978
// MI455X (gfx1250) — compile-verified
//
#include <hip/hip_runtime.h>
#include <hip/hip_bf16.h>

// ---------------------------------------------------------------------------
// GATv2 (2 layers, H=2, C=64) + GraphNorm + MLP head, fp32 end-to-end.
// Dense GEMMs use CDNA5 V_WMMA_F32_16X16X4_F32 (exact fp32, wave32).
// ---------------------------------------------------------------------------

#define GN 50000        // nodes
#define GE 800000       // edges (before self-loops)
#define GETOT (GE + GN) // edges incl. self-loops
#define GHC 128         // H*C
#define GC 64
#define GH 2
#define NEG_SLOPE 0.2f
#define GEPS 1e-5f

typedef __attribute__((ext_vector_type(2))) float v2f;
typedef __attribute__((ext_vector_type(8))) float v8f;

// ---------------------------------------------------------------------------
// fp32 WMMA GEMM:  C[M x Nout] = A[M x K] @ B[K x Nout] (+bias, +relu)
// One wave computes one 16x16 tile. blockDim = (32, Nout/16), grid.x = M/16.
// A-frag (16x4 f32): lanes 0-15 -> M=lane, K=kk+0,kk+1 ; lanes 16-31 -> K=kk+2,kk+3
// B-frag (4x16 f32): lanes 0-15 -> K=kk+0,kk+1 ; lanes 16-31 -> K=kk+2,kk+3 ; N=lane&15
// C/D   (16x16 f32): VGPR j: lanes 0-15 -> row j, lanes 16-31 -> row j+8 ; N=lane&15
// ---------------------------------------------------------------------------
__global__ void wmma_gemm_f32(const float* __restrict__ A,
                              const float* __restrict__ B,
                              const float* __restrict__ bias,
                              float* __restrict__ C,
                              int K, int Nout, int doRelu) {
  const int lane    = threadIdx.x;            // 0..31
  const int colTile = threadIdx.y;            // 0..Nout/16-1
  const int row0    = blockIdx.x * 16;
  const int col0    = colTile * 16;
  const int col     = lane & 15;
  const int kHalf   = (lane >> 4) << 1;       // 0 or 2
  const int aRow    = row0 + (lane & 15);

  v8f acc = {};
  for (int kk = 0; kk < K; kk += 4) {
    v2f a, b;
    const float* ap = A + (size_t)aRow * K + (kk + kHalf);
    a.x = ap[0];
    a.y = ap[1];
    const float* bp = B + (size_t)(kk + kHalf) * Nout + col0 + col;
    b.x = bp[0];
    b.y = bp[Nout];
    acc = __builtin_amdgcn_wmma_f32_16x16x4_f32(false, a, false, b,
                                                (short)0, acc, false, false);
  }

  const int rowBase = row0 + ((lane >> 4) << 3);   // +0 or +8
  const float bv = bias ? bias[col0 + col] : 0.0f;
#pragma unroll
  for (int j = 0; j < 8; ++j) {
    float v = acc[j] + bv;
    if (doRelu) v = fmaxf(v, 0.0f);
    C[(size_t)(rowBase + j) * Nout + col0 + col] = v;
  }
}

// ---------------------------------------------------------------------------
// Utility fills
// ---------------------------------------------------------------------------
__global__ void fill_kernel(float* __restrict__ p, float v, int n) {
  int i = blockIdx.x * blockDim.x + threadIdx.x;
  if (i < n) p[i] = v;
}

// Sign-aware float atomic max via int/uint atomics (works for mixed signs).
__device__ inline void atomicMaxF(float* addr, float val) {
  if (val >= 0.0f)
    atomicMax((int*)addr, __float_as_int(val));
  else
    atomicMin((unsigned int*)addr, __float_as_uint(val));
}

// ---------------------------------------------------------------------------
// Edge phase 1: logits[e,h] = sum_c att[h,c] * leaky_relu(xl[src,h,c]+xr[dst,h,c])
// one wave (32 lanes) per (edge, head); lane handles channels c and c+32.
// Also accumulates per-(dst,head) running max (stable softmax).
// ---------------------------------------------------------------------------
__global__ void edge_logits_kernel(const float* __restrict__ xl,
                                   const float* __restrict__ xr,
                                   const int* __restrict__ ei,
                                   const float* __restrict__ att,
                                   float* __restrict__ logits,
                                   float* __restrict__ m) {
  const int lane = threadIdx.x & 31;
  const int wave = threadIdx.x >> 5;
  const int pair = blockIdx.x * (blockDim.x >> 5) + wave;   // wave-uniform
  if (pair >= GETOT * GH) return;                           // whole wave exits
  const int e = pair >> 1;
  const int h = pair & 1;
  const int s = (e < GE) ? ei[e] : (e - GE);
  const int d = (e < GE) ? ei[GE + e] : (e - GE);

  const float* xls = xl + (size_t)s * GHC + h * GC;
  const float* xrd = xr + (size_t)d * GHC + h * GC;
  const float* ah  = att + h * GC;

  float t0 = xls[lane] + xrd[lane];
  float t1 = xls[lane + 32] + xrd[lane + 32];
  t0 = (t0 > 0.0f) ? t0 : NEG_SLOPE * t0;
  t1 = (t1 > 0.0f) ? t1 : NEG_SLOPE * t1;
  float v = t0 * ah[lane] + t1 * ah[lane + 32];

#pragma unroll
  for (int off = 16; off > 0; off >>= 1) v += __shfl_xor(v, off, 32);

  if (lane == 0) {
    logits[pair] = v;
    atomicMaxF(&m[(size_t)d * GH + h], v);
  }
}

// ---------------------------------------------------------------------------
// Edge phase 2: p = exp(logit - m[dst]); denom[dst] += p (in place over logits)
// ---------------------------------------------------------------------------
__global__ void edge_softmax_p(const int* __restrict__ ei,
                               float* __restrict__ logits,
                               const float* __restrict__ m,
                               float* __restrict__ denom) {
  int i = blockIdx.x * blockDim.x + threadIdx.x;
  if (i >= GETOT * GH) return;
  const int e = i >> 1;
  const int h = i & 1;
  const int d = (e < GE) ? ei[GE + e] : (e - GE);
  float p = __expf(logits[i] - m[(size_t)d * GH + h]);
  logits[i] = p;
  atomicAdd(&denom[(size_t)d * GH + h], p);
}

// ---------------------------------------------------------------------------
// Edge phase 3: accum[dst, f] += (p/denom[dst]) * xl[src, f]
// one 128-thread block per edge (thread f = feature, head = f>>6)
// ---------------------------------------------------------------------------
__global__ void edge_aggregate(const int* __restrict__ ei,
                               const float* __restrict__ xl,
                               const float* __restrict__ p,
                               const float* __restrict__ denom,
                               float* __restrict__ accum) {
  const int e = blockIdx.x;
  const int f = threadIdx.x;            // 0..127
  const int h = f >> 6;
  const int s = (e < GE) ? ei[e] : (e - GE);
  const int d = (e < GE) ? ei[GE + e] : (e - GE);
  const float alpha = p[(size_t)e * GH + h] / denom[(size_t)d * GH + h];
  atomicAdd(&accum[(size_t)d * GHC + f], alpha * xl[(size_t)s * GHC + f]);
}

// ---------------------------------------------------------------------------
// GraphNorm stats: per-column sum & sumsq of (accum + conv_bias), 128 rows/block
// ---------------------------------------------------------------------------
__global__ void gn_stats(const float* __restrict__ accum,
                         const float* __restrict__ bias,
                         float* __restrict__ colsum,
                         float* __restrict__ colsumsq) {
  const int f = threadIdx.x;            // 0..127
  const int row0 = blockIdx.x * 128;
  const float bv = bias[f];
  float s = 0.0f, s2 = 0.0f;
  for (int r = 0; r < 128; ++r) {
    const int n = row0 + r;
    if (n >= GN) break;
    const float x = accum[(size_t)n * GHC + f] + bv;
    s += x;
    s2 += x * x;
  }
  atomicAdd(&colsum[f], s);
  atomicAdd(&colsumsq[f], s2);
}

// var(o) with o = x - ms*mean :  E[x^2] - (2*ms - ms^2) * mean^2
// out = w*o*rsqrt(var+eps)+b  ->  x*s1 + s2 with s1=w*r, s2=b - s1*ms*mean
__global__ void gn_finalize(const float* __restrict__ colsum,
                            const float* __restrict__ colsumsq,
                            const float* __restrict__ w,
                            const float* __restrict__ b,
                            const float* __restrict__ ms,
                            float* __restrict__ s1,
                            float* __restrict__ s2) {
  const int f = threadIdx.x;            // 0..127
  const float invN = 1.0f / (float)GN;
  const float mean = colsum[f] * invN;
  const float ex2  = colsumsq[f] * invN;
  const float msv  = ms[f];
  const float var  = ex2 - (2.0f * msv - msv * msv) * mean * mean;
  const float r    = rsqrtf(var + GEPS);
  const float a    = w[f] * r;
  s1[f] = a;
  s2[f] = b[f] - a * msv * mean;
}

__global__ void gn_apply(const float* __restrict__ accum,
                         const float* __restrict__ bias,
                         const float* __restrict__ s1,
                         const float* __restrict__ s2,
                         float* __restrict__ out) {
  int i = blockIdx.x * blockDim.x + threadIdx.x;
  if (i >= GN * GHC) return;
  const int f = i & (GHC - 1);
  const float x = accum[i] + bias[f];
  out[i] = fmaxf(x * s1[f] + s2[f], 0.0f);
}

// ---------------------------------------------------------------------------
// Head: out[n, 0:2] = h1[n, :] @ W2 + b2   (W2: 64x2, tiny -> plain VALU)
// ---------------------------------------------------------------------------
__global__ void head_kernel(const float* __restrict__ h1,
                            const float* __restrict__ W2,
                            const float* __restrict__ b2,
                            float* __restrict__ out) {
  int n = blockIdx.x * blockDim.x + threadIdx.x;
  if (n >= GN) return;
  const float* row = h1 + (size_t)n * GC;
  float a0 = b2[0], a1 = b2[1];
#pragma unroll 8
  for (int c = 0; c < GC; ++c) {
    const float v = row[c];
    a0 += v * W2[c * 2 + 0];
    a1 += v * W2[c * 2 + 1];
  }
  out[n * 2 + 0] = a0;
  out[n * 2 + 1] = a1;
}

// ---------------------------------------------------------------------------
// Orchestration
// ---------------------------------------------------------------------------
extern "C" void kernel_launch(void* const* d_in, const int* in_sizes, int n_in,
                              void* d_out, int out_size, void* d_ws, size_t ws_size,
                              hipStream_t stream) {
  (void)in_sizes; (void)n_in; (void)out_size; (void)ws_size;

  const float* x     = (const float*)d_in[0];
  const int*   ei    = (const int*)d_in[1];
  const float* Wl    = (const float*)d_in[2];   // [2,128,128]
  const float* Wr    = (const float*)d_in[3];   // [2,128,128]
  const float* att   = (const float*)d_in[4];   // [2,2,64]
  const float* convb = (const float*)d_in[5];   // [2,128]
  const float* gnw   = (const float*)d_in[6];
  const float* gnb   = (const float*)d_in[7];
  const float* gnms  = (const float*)d_in[8];
  const float* W1    = (const float*)d_in[9];   // [128,64]
  const float* b1    = (const float*)d_in[10];  // [64]
  const float* W2    = (const float*)d_in[11];  // [64,2]
  const float* b2    = (const float*)d_in[12];  // [2]
  float* out = (float*)d_out;

  // Workspace layout (floats). logits aliases h1 (disjoint lifetimes).
  float* ws       = (float*)d_ws;
  float* xl       = ws;                          // GN*GHC
  float* xr       = xl + (size_t)GN * GHC;       // GN*GHC
  float* hbuf     = xr + (size_t)GN * GHC;       // GN*GHC
  float* h1       = hbuf + (size_t)GN * GHC;     // max(GN*GC, GETOT*GH)
  float* logits   = h1;                          //   alias
  size_t scratch1 = ((size_t)GN * GC > (size_t)GETOT * GH)
                        ? (size_t)GN * GC : (size_t)GETOT * GH;
  float* accum    = h1 + scratch1;               // GN*GHC   } zeroed
  float* denom    = accum + (size_t)GN * GHC;    // GN*GH    } per
  float* colsum   = denom + (size_t)GN * GH;     // 128      } layer
  float* colsumsq = colsum + GHC;                // 128      }
  float* s1       = colsumsq + GHC;              // 128
  float* s2       = s1 + GHC;                    // 128
  float* m        = s2 + GHC;                    // GN*GH (filled with -3e38)

  const int zeroCount = GN * GHC + GN * GH + 2 * GHC;  // accum..colsumsq
  const int pairCount = GETOT * GH;

  const float* cur = x;
  for (int layer = 0; layer < 2; ++layer) {
    const float* wl = Wl + (size_t)layer * GHC * GHC;
    const float* wr = Wr + (size_t)layer * GHC * GHC;
    const float* at = att + (size_t)layer * GH * GC;
    const float* cb = convb + (size_t)layer * GHC;

    // xl = cur @ Wl, xr = cur @ Wr  (WMMA fp32)
    wmma_gemm_f32<<<GN / 16, dim3(32, GHC / 16), 0, stream>>>(cur, wl, nullptr, xl, GHC, GHC, 0);
    wmma_gemm_f32<<<GN / 16, dim3(32, GHC / 16), 0, stream>>>(cur, wr, nullptr, xr, GHC, GHC, 0);

    // init accumulators
    fill_kernel<<<(zeroCount + 255) / 256, 256, 0, stream>>>(accum, 0.0f, zeroCount);
    fill_kernel<<<(GN * GH + 255) / 256, 256, 0, stream>>>(m, -3.0e38f, GN * GH);

    // edge attention
    edge_logits_kernel<<<(pairCount + 7) / 8, 256, 0, stream>>>(xl, xr, ei, at, logits, m);
    edge_softmax_p<<<(pairCount + 255) / 256, 256, 0, stream>>>(ei, logits, m, denom);
    edge_aggregate<<<GETOT, GHC, 0, stream>>>(ei, xl, logits, denom, accum);

    // GraphNorm + ReLU -> hbuf
    gn_stats<<<(GN + 127) / 128, GHC, 0, stream>>>(accum, cb, colsum, colsumsq);
    gn_finalize<<<1, GHC, 0, stream>>>(colsum, colsumsq,
                                       gnw + (size_t)layer * GHC,
                                       gnb + (size_t)layer * GHC,
                                       gnms + (size_t)layer * GHC, s1, s2);
    gn_apply<<<(GN * GHC + 255) / 256, 256, 0, stream>>>(accum, cb, s1, s2, hbuf);

    cur = hbuf;
  }

  // head: h1 = relu(hbuf @ W1 + b1) ; out = h1 @ W2 + b2
  wmma_gemm_f32<<<GN / 16, dim3(32, GC / 16), 0, stream>>>(hbuf, W1, b1, h1, GHC, GC, 1);
  head_kernel<<<(GN + 255) / 256, 256, 0, stream>>>(h1, W2, b2, out);
}